// SpectralWindowPreprocessor_26912265076910
// MI455X (gfx1250) — compile-verified
//
#include <hip/hip_runtime.h>
#include <hip/hip_bf16.h>

// SpectralWindowPreprocessor for gfx1250 (MI455X).
// out[b,c,j,h,w] = x[b, reflect(c + j - 3), h, w]   (mask == 1 for 'reflect')
//
// Memory-bound: 130 MB read (once, via TDM into LDS) + 910 MB written
// (non-temporal) => ~45us floor @ 23.3 TB/s. The Tensor Data Mover loads a
// [31 ch x 1024 px] 2D tile (row stride = HW) into 124 KB of LDS with a
// single tensor_load_to_lds issued by wave 0; completion is tracked with
// TENSORcnt and published to the workgroup with the split barrier. The 7x
// channel-window broadcast is then served from LDS (ds_load_b128) and
// streamed out with th:TH_STORE_NT b128 stores so the write stream does not
// thrash the 192 MB L2.

#define C_CH   31
#define WIN    7
#define HALF   3
#define IMG_HW (512 * 512)
#define BATCH  4
#define BLOCK  256
#define VEC    4                    // floats per lane per access (B128)
#define TILE_P (BLOCK * VEC)        // 1024 pixels per tile per channel

typedef float    v4f __attribute__((ext_vector_type(4)));
typedef uint32_t u32;
typedef u32      v4u __attribute__((ext_vector_type(4)));
typedef u32      v8u __attribute__((ext_vector_type(8)));

__global__ __launch_bounds__(BLOCK) void spectral_window_gfx1250(
    const float* __restrict__ x, float* __restrict__ out) {
  __shared__ float smem[C_CH][TILE_P];   // 31 * 4 KB = 124 KB LDS

  const int tid = threadIdx.x;
  const int b   = blockIdx.x >> 8;              // 256 tiles per image
  const int p0  = (int)(blockIdx.x & 255) * TILE_P;

  // ---------------- TDM fill: one 2D tensor tile -> LDS ---------------------
  // Tile: dim1 (rows)   = 31 channels, row stride = IMG_HW elements
  //       dim0 (cols)   = 1024 contiguous f32 pixels
  // Lands packed into smem[31][1024].
  const float*   gtile    = x + (size_t)b * (size_t)C_CH * (size_t)IMG_HW + (size_t)p0;
  const uint64_t gaddr    = (uint64_t)(uintptr_t)gtile;
  const u32      lds_base = (u32)(size_t)(&smem[0][0]);  // wave-relative LDS byte addr

  // D# group 0: count=1 | lds_addr | global_addr[56:0] | type=2
  v4u g0 = { 1u,
             lds_base,
             (u32)gaddr,
             ((u32)(gaddr >> 32) & 0x01FFFFFFu) | 0x80000000u };

  // D# group 1: mask=0, data_size=2 (4B), no pad/iterate/gather,
  //             tensor_dim0=1024, tensor_dim1=31, tile_dim0=1024, tile_dim1=31,
  //             tile_dim2=0, tensor_dim0_stride=IMG_HW, dim1_stride unused.
  v8u g1 = { 0x00020000u,                                   // data_size=4B
             ((u32)TILE_P & 0xFFFFu) << 16,                 // tensor_dim0 lo16
             ((u32)TILE_P >> 16) | (((u32)C_CH & 0xFFFFu) << 16), // dim0 hi | dim1 lo
             ((u32)C_CH >> 16) | (((u32)TILE_P & 0xFFFFu) << 16), // dim1 hi | tile_dim0
             (u32)C_CH,                                     // tile_dim1=31, tile_dim2=0
             (u32)IMG_HW,                                   // tensor_dim0_stride lo32
             0u,                                            // stride hi16 | dim1_stride lo
             0u };

  // Only wave 0 issues the DMA (TDM ignores EXEC, so the guard must be a true
  // scalar branch -> uniformize with readfirstlane).
  const int wid = __builtin_amdgcn_readfirstlane((int)threadIdx.x >> 5);
  if (wid == 0) {
    asm volatile("tensor_load_to_lds %0, %1"
                 :
                 : "s"(g0), "s"(g1)
                 : "memory");
    asm volatile("s_wait_tensorcnt 0x0" ::: "memory");
  }
  __syncthreads();   // publish the 124 KB tile to all 8 waves

  // ---------------- gather + 7x non-temporal streaming stores --------------
  float* obase = out + (size_t)b * (size_t)C_CH * (size_t)WIN * (size_t)IMG_HW
                     + (size_t)p0 + (size_t)(tid * VEC);

  #pragma unroll 1
  for (int c = 0; c < C_CH; ++c) {
    float* op = obase + (size_t)c * (size_t)WIN * (size_t)IMG_HW;
    #pragma unroll
    for (int j = 0; j < WIN; ++j) {
      int t = c + j - HALF;                       // reflect padding (uniform)
      t = (t < 0)     ? (-t - 1)           : t;
      t = (t >= C_CH) ? (2 * C_CH - 1 - t) : t;
      v4f v = *(const v4f*)(&smem[t][tid * VEC]); // ds_load_b128
      __builtin_nontemporal_store(v, (v4f*)(op + (size_t)j * IMG_HW)); // b128 NT
    }
  }
}

extern "C" void kernel_launch(void* const* d_in, const int* in_sizes, int n_in,
                              void* d_out, int out_size, void* d_ws, size_t ws_size,
                              hipStream_t stream) {
  const float* x   = (const float*)d_in[0];
  float*       out = (float*)d_out;

  dim3 grid(BATCH * (IMG_HW / TILE_P));  // 4 * 256 = 1024 workgroups
  dim3 block(BLOCK);
  hipLaunchKernelGGL(spectral_window_gfx1250, grid, block, 0, stream, x, out);
}